// MixQuantActiv_87617332839035
// MI455X (gfx1250) — compile-verified
//
#include <hip/hip_runtime.h>

// Problem geometry (compile-time, from the reference)
#define BB     32
#define CCH    256
#define HW     3136            // 56*56
#define HW4    784             // float4 per (b,c) plane
#define NSEL   8
#define NPLANE (BB * NSEL)     // 256 selected planes
#define SEL4   (NPLANE * HW4)  // 200704 float4 of gathered data = 784 blocks * 256 thr
#define N4     (BB * CCH * HW4)// 6,422,528 float4 total = 3136 tiles * 2048 float4
#define NFLT   (N4 * 4)        // 25,690,112 floats in result tensor

// d_ws layout (float indices)
#define WS_MIN 0
#define WS_MAX 1024
#define WS_PAR 2048            // [0]=mn [1..3]=inv_scale(2,4,8b) [4..6]=sw_i*scale_i

typedef float v4f __attribute__((ext_vector_type(4)));

// ---------------- Kernel 3 (placed first so its disasm shows in the snippet):
// bulk copy via the gfx1250 async global<->LDS data mover.
// 3136 blocks x 256 threads; one 32KB tile/block (2048 float4), exact cover of N4.
// Each wave loads/stores only its own LDS slots -> no barrier needed; the load
// phase keeps 8 B128 async ops in flight per lane-slot, one ASYNCcnt turnaround,
// then 8 async stores; s_endpgm's implicit wait-idle retires them. NT hints keep
// the 206MB one-shot stream from evicting the selected planes out of L2.
__global__ __launch_bounds__(256) void k_copy_async(
    const float* __restrict__ in, float* __restrict__ out) {
  __shared__ __align__(16) unsigned char smem[32768];
  const unsigned lbase = (unsigned)(unsigned long long)(smem);  // low 32b = LDS byte offset
  const unsigned t     = threadIdx.x;
  const unsigned gbase = blockIdx.x * (2048u * 16u);            // byte offset of tile
#pragma unroll
  for (int k = 0; k < 8; ++k) {
    const unsigned e = (unsigned)k * 256u + t;
    asm volatile("global_load_async_to_lds_b128 %0, %1, %2 th:TH_LOAD_NT"
                 :: "v"(lbase + e * 16u), "v"(gbase + e * 16u), "s"(in)
                 : "memory");
  }
  asm volatile("s_wait_asynccnt 0" ::: "memory");
#pragma unroll
  for (int k = 0; k < 8; ++k) {
    const unsigned e = (unsigned)k * 256u + t;
    asm volatile("global_store_async_from_lds_b128 %0, %1, %2 th:TH_STORE_NT"
                 :: "v"(gbase + e * 16u), "v"(lbase + e * 16u), "s"(out)
                 : "memory");
  }
}

// ---------------- Kernel 1: per-block min/max over gathered channels ----------
__global__ __launch_bounds__(256) void k_minmax_part(
    const float* __restrict__ in, const int* __restrict__ sel, float* __restrict__ ws) {
  const int t  = threadIdx.x;
  const int i4 = blockIdx.x * 256 + t;          // < SEL4 exactly
  const int plane = i4 / HW4;
  const int hw4   = i4 - plane * HW4;
  const int b = plane >> 3, s = plane & 7;
  const int c = sel[s];
  const v4f v = ((const v4f*)in)[(size_t)(b * CCH + c) * HW4 + hw4];

  float mn = fminf(fminf(v[0], v[1]), fminf(v[2], v[3]));
  float mx = fmaxf(fmaxf(v[0], v[1]), fmaxf(v[2], v[3]));
#pragma unroll
  for (int off = 16; off >= 1; off >>= 1) {     // wave32 tree reduce
    mn = fminf(mn, __shfl_xor(mn, off, 32));
    mx = fmaxf(mx, __shfl_xor(mx, off, 32));
  }
  __shared__ float smn[8], smx[8];
  if ((t & 31) == 0) { smn[t >> 5] = mn; smx[t >> 5] = mx; }
  __syncthreads();
  if (t == 0) {
#pragma unroll
    for (int w = 1; w < 8; ++w) { mn = fminf(mn, smn[w]); mx = fmaxf(mx, smx[w]); }
    ws[WS_MIN + blockIdx.x] = mn;
    ws[WS_MAX + blockIdx.x] = mx;
  }
}

// ---------------- Kernel 2: fold partials, softmax(beta), emit params ---------
__global__ __launch_bounds__(256) void k_finalize(
    const float* __restrict__ beta, float* __restrict__ ws, float* __restrict__ out_scale) {
  const int t = threadIdx.x;
  float mn = 3.402823466e38f, mx = -3.402823466e38f;
  for (int j = t; j < 784; j += 256) {
    mn = fminf(mn, ws[WS_MIN + j]);
    mx = fmaxf(mx, ws[WS_MAX + j]);
  }
#pragma unroll
  for (int off = 16; off >= 1; off >>= 1) {
    mn = fminf(mn, __shfl_xor(mn, off, 32));
    mx = fmaxf(mx, __shfl_xor(mx, off, 32));
  }
  __shared__ float smn[8], smx[8];
  if ((t & 31) == 0) { smn[t >> 5] = mn; smx[t >> 5] = mx; }
  __syncthreads();
  if (t == 0) {
#pragma unroll
    for (int w = 1; w < 8; ++w) { mn = fminf(mn, smn[w]); mx = fmaxf(mx, smx[w]); }
    const float b0 = beta[0], b1 = beta[1], b2 = beta[2];
    const float m  = fmaxf(b0, fmaxf(b1, b2));
    const float e0 = __expf(b0 - m), e1 = __expf(b1 - m), e2 = __expf(b2 - m);
    const float r  = 1.0f / (e0 + e1 + e2);
    const float sw[3]   = {e0 * r, e1 * r, e2 * r};
    const float qmax[3] = {3.0f, 15.0f, 255.0f};
    const float range   = mx - mn;
    ws[WS_PAR + 0] = mn;
    float last = 0.0f;
#pragma unroll
    for (int i = 0; i < 3; ++i) {
      const float sc = range / qmax[i];
      ws[WS_PAR + 1 + i] = qmax[i] / range;   // 1/scale_i
      ws[WS_PAR + 4 + i] = sw[i] * sc;        // sw_i * scale_i
      last = sc;
    }
    *out_scale = last;                        // scale from bit==8 (last iteration)
  }
}

// ---------------- Kernel 4: quantize-blend the selected planes ----------------
__global__ __launch_bounds__(256) void k_fixup(
    const float* __restrict__ in, const int* __restrict__ sel,
    const float* __restrict__ ws, float* __restrict__ out) {
  const int t  = threadIdx.x;
  const int i4 = blockIdx.x * 256 + t;          // < SEL4 exactly
  const int plane = i4 / HW4;
  const int hw4   = i4 - plane * HW4;
  const int b = plane >> 3, s = plane & 7;
  const int c = sel[s];
  const size_t addr = (size_t)(b * CCH + c) * HW4 + hw4;

  const v4f v = ((const v4f*)in)[addr];
  const float mn = ws[WS_PAR + 0];
  const float i0 = ws[WS_PAR + 1], i1 = ws[WS_PAR + 2], i2 = ws[WS_PAR + 3];
  const float w0 = ws[WS_PAR + 4], w1 = ws[WS_PAR + 5], w2 = ws[WS_PAR + 6];

  v4f r;
#pragma unroll
  for (int j = 0; j < 4; ++j) {
    const float d = v[j] - mn;                  // RNE rint == jnp.round
    float acc = mn;                             // sum(sw)==1 -> mn term folds out
    acc = fmaf(fminf(fmaxf(__builtin_rintf(d * i0), 0.0f),   3.0f), w0, acc);
    acc = fmaf(fminf(fmaxf(__builtin_rintf(d * i1), 0.0f),  15.0f), w1, acc);
    acc = fmaf(fminf(fmaxf(__builtin_rintf(d * i2), 0.0f), 255.0f), w2, acc);
    r[j] = acc;
  }
  __builtin_nontemporal_store(r, (v4f*)out + addr);   // write-once, bypass near cache
}

// ---------------- Host launcher ----------------------------------------------
extern "C" void kernel_launch(void* const* d_in, const int* in_sizes, int n_in,
                              void* d_out, int out_size, void* d_ws, size_t ws_size,
                              hipStream_t stream) {
  const float* in   = (const float*)d_in[0];
  const float* beta = (const float*)d_in[1];
  const int*   sel  = (const int*)d_in[2];
  float* out = (float*)d_out;
  float* ws  = (float*)d_ws;

  k_minmax_part<<<784, 256, 0, stream>>>(in, sel, ws);
  k_finalize  <<<1,   256, 0, stream>>>(beta, ws, out + NFLT);
  k_copy_async<<<3136, 256, 0, stream>>>(in, out);
  k_fixup     <<<784, 256, 0, stream>>>(in, sel, ws, out);
}